// HungarianL1Loss_2044404433129
// MI455X (gfx1250) — compile-verified
//
#include <hip/hip_runtime.h>

typedef __attribute__((ext_vector_type(2))) float v2f;
typedef __attribute__((ext_vector_type(8))) float v8f;

#define NPTS 64

// One wave32 per batch problem. Jonker-Volgenant shortest augmenting path,
// columns striped 2-per-lane, cost matrix built with V_WMMA_F32_16X16X4_F32.
__global__ __launch_bounds__(32) void hungarian_block(const float* __restrict__ pred,
                                                      const float* __restrict__ target,
                                                      float* __restrict__ block_sums)
{
    __shared__ float sCost[NPTS * NPTS];   // 16 KB: cost[i*64 + j], rows=target, cols=pred
    __shared__ float sU[NPTS + 1];         // row duals, 1-indexed (u[0] virtual)
    __shared__ int   sP[NPTS + 1];         // p[j] = row matched to column j (1-indexed)
    __shared__ int   sWay[NPTS + 1];       // augmenting-path predecessor column

    const int tid  = threadIdx.x;          // 0..31
    const int b    = blockIdx.x;           // 0..127
    const int lane = tid;
    const bool lo  = lane < 16;

    const float* predB = pred   + (size_t)b * NPTS * 2;
    const float* tgtB  = target + (size_t)b * NPTS * 2;

    // ---------------- Cost matrix via WMMA ----------------
    // D[i][j] = t_i - p_j = A x B,  A row i = [t_i, 1, 0, 0],  B col j = [1, -p_j, 0, 0]^T
    // A 16x4 f32 layout: lanes 0-15 hold K=0 (vgpr0), K=1 (vgpr1); lanes 16-31 hold K=2,3 (zeroed).
    for (int ti = 0; ti < 4; ++ti) {
        const int rowIdx = ti * 16 + (lane & 15);
        const float tx = tgtB[rowIdx * 2 + 0];
        const float ty = tgtB[rowIdx * 2 + 1];
        v2f ax, ay;
        ax[0] = lo ? tx : 0.f;  ax[1] = lo ? 1.f : 0.f;
        ay[0] = lo ? ty : 0.f;  ay[1] = lo ? 1.f : 0.f;
        for (int tj = 0; tj < 4; ++tj) {
            const int colIdx = tj * 16 + (lane & 15);
            const float px = predB[colIdx * 2 + 0];
            const float py = predB[colIdx * 2 + 1];
            v2f bx, by;
            bx[0] = lo ? 1.f : 0.f;  bx[1] = lo ? -px : 0.f;
            by[0] = lo ? 1.f : 0.f;  by[1] = lo ? -py : 0.f;
            v8f cz = {};
            v8f dx = __builtin_amdgcn_wmma_f32_16x16x4_f32(false, ax, false, bx,
                                                           (short)0, cz, false, false);
            v8f dy = __builtin_amdgcn_wmma_f32_16x16x4_f32(false, ay, false, by,
                                                           (short)0, cz, false, false);
            // C/D layout: VGPR r -> row (r | r+8), col = lane&15
            const int rbase = ti * 16 + (lo ? 0 : 8);
            const int col   = tj * 16 + (lane & 15);
#pragma unroll
            for (int r = 0; r < 8; ++r)
                sCost[(rbase + r) * NPTS + col] = fabsf(dx[r]) + fabsf(dy[r]);
        }
    }

    // init duals / matching (65 entries, 32 lanes)
    sU[tid] = 0.f;  sU[tid + 32] = 0.f;
    sP[tid] = 0;    sP[tid + 32] = 0;
    sWay[tid] = 0;  sWay[tid + 32] = 0;
    if (tid == 0) { sU[64] = 0.f; sP[64] = 0; sWay[64] = 0; }
    __syncthreads();

    const int jA = tid + 1, jB = tid + 33;   // owned columns (1-indexed)
    float vA = 0.f, vB = 0.f;                // column duals

    for (int i = 1; i <= NPTS; ++i) {
        if (tid == 0) sP[0] = i;
        __syncthreads();

        float minvA = __builtin_inff(), minvB = __builtin_inff();
        bool usedA = false, usedB = false;
        int j0 = 0;

        while (true) {
            if (jA == j0) usedA = true;
            if (jB == j0) usedB = true;
            const int   i0  = sP[j0];
            const float ui0 = sU[i0];
            const float* crow = &sCost[(i0 - 1) * NPTS];

            if (!usedA) {
                float cur = crow[jA - 1] - ui0 - vA;
                if (cur < minvA) { minvA = cur; sWay[jA] = j0; }
            }
            if (!usedB) {
                float cur = crow[jB - 1] - ui0 - vB;
                if (cur < minvB) { minvB = cur; sWay[jB] = j0; }
            }

            // wave argmin over free columns (tie -> smallest column index, like np.argmin)
            float va = usedA ? __builtin_inff() : minvA;
            float vb = usedB ? __builtin_inff() : minvB;
            float val; int idx;
            if (vb < va) { val = vb; idx = jB; } else { val = va; idx = jA; }
#pragma unroll
            for (int off = 16; off > 0; off >>= 1) {
                float ov = __shfl_xor(val, off, 32);
                int   oi = __shfl_xor(idx, off, 32);
                if (ov < val || (ov == val && oi < idx)) { val = ov; idx = oi; }
            }
            const float delta = val;
            const int   j1    = idx;

            __syncthreads();
            // dual updates: used rows are pairwise-distinct -> race-free LDS RMW
            if (tid == 0) sU[i] += delta;                 // column 0 (always used), p[0]=i
            if (usedA) { int r = sP[jA]; sU[r] += delta; vA -= delta; }
            else       { minvA -= delta; }
            if (usedB) { int r = sP[jB]; sU[r] += delta; vB -= delta; }
            else       { minvB -= delta; }
            __syncthreads();

            j0 = j1;
            if (sP[j0] == 0) break;
        }

        // augment along predecessor chain
        while (true) {
            const int j1a = sWay[j0];
            const int pj1 = sP[j1a];
            __syncthreads();
            if (tid == 0) sP[j0] = pj1;
            __syncthreads();
            j0 = j1a;
            if (j0 == 0) break;
        }
    }

    // sum of matched costs: sum_j cost[p[j]-1][j-1]
    __syncthreads();
    float s = sCost[(sP[jA] - 1) * NPTS + (jA - 1)]
            + sCost[(sP[jB] - 1) * NPTS + (jB - 1)];
#pragma unroll
    for (int off = 16; off > 0; off >>= 1) s += __shfl_xor(s, off, 32);
    if (tid == 0) block_sums[b] = s;
}

__global__ __launch_bounds__(32) void reduce_final(const float* __restrict__ block_sums,
                                                   float* __restrict__ out)
{
    const int tid = threadIdx.x;
    float s = block_sums[tid] + block_sums[tid + 32] +
              block_sums[tid + 64] + block_sums[tid + 96];
#pragma unroll
    for (int off = 16; off > 0; off >>= 1) s += __shfl_xor(s, off, 32);
    if (tid == 0) out[0] = s * (1.0f / (128.0f * 64.0f * 2.0f));
}

extern "C" void kernel_launch(void* const* d_in, const int* in_sizes, int n_in,
                              void* d_out, int out_size, void* d_ws, size_t ws_size,
                              hipStream_t stream) {
    const float* pred   = (const float*)d_in[0];   // [128,64,2] f32
    const float* target = (const float*)d_in[1];   // [128,64,2] f32
    float* bsums = (float*)d_ws;                   // 128 floats scratch (fully overwritten)
    hungarian_block<<<128, 32, 0, stream>>>(pred, target, bsums);
    reduce_final<<<1, 32, 0, stream>>>(bsums, (float*)d_out);
}